// RNNLayer_2345052143928
// MI455X (gfx1250) — compile-verified
//
#include <hip/hip_runtime.h>
#include <cstdint>
#include <cstddef>

typedef __attribute__((ext_vector_type(2))) float v2f;
typedef __attribute__((ext_vector_type(8))) float v8f;

#define B_  512
#define T_  32
#define D_  512
#define U_  512
#define O_  512

#define BM  64      // rows (batch) per block
#define BN  128     // cols per block
#define KC  32      // K chunk staged in LDS
#define LDA 34      // padded LDS stride for A chunk (even -> b64 aligned)
#define LDBP 264    // padded LDS stride for packed B pair-rows (2*BN + 8, even)
#define NTHR 256    // 8 wave32

// Deterministic pseudo-dropout (keep=0.8, scale 1/0.8). JAX threefry is not
// reproducible on-device; this is a deterministic stand-in.
__device__ __forceinline__ float drop_mask(uint32_t salt, uint32_t t,
                                           uint32_t r, uint32_t c) {
  uint32_t h = salt * 0x9E3779B9u;
  h ^= t * 0x85EBCA6Bu + (h << 6) + (h >> 2);
  h ^= r * 0xC2B2AE35u;
  h ^= c * 0x27D4EB2Fu;
  h ^= h >> 16; h *= 0x7FEB352Du;
  h ^= h >> 15; h *= 0x846CA68Bu;
  h ^= h >> 16;
  return (h < 0xCCCCCCCCu) ? 1.25f : 0.0f;  // P(keep)=0.8, scale 1/0.8
}

// Accumulate acc[2][2] += A[64 x K] * W[K x 512](cols colBase..colBase+127).
// A rows are global batch rows rowBase..rowBase+63 with row stride arow.
// maskSalt != 0 applies dropout mask to A elements.
// LDS B layout is K-pair interleaved: element (k, n) lives at
//   lB[(k>>1)*LDBP + 2*n + (k&1)]
// so a lane's B fragment {K=lk2, K=lk2+1} at column n is one aligned b64.
__device__ __forceinline__ void gemm_segment(
    const float* __restrict__ A, size_t arow,
    const float* __restrict__ W, int K,
    int rowBase, int colBase, uint32_t maskSalt, int t,
    float* __restrict__ lA, float* __restrict__ lB,
    v8f acc[2][2], int wr, int wc, int lane) {
  const int tid = threadIdx.x;
  const int lm  = lane & 15;
  const int lk2 = (lane >> 4) * 2;
  const int m0  = wr * 32;
  const int n0  = wc * 32;

  for (int k0 = 0; k0 < K; k0 += KC) {
    __syncthreads();  // protect LDS from previous chunk's readers
    // ---- stage A chunk: 64 rows x 32 k (8 floats per thread) ----
    {
      int r  = tid >> 2;             // 0..63
      int kk = (tid & 3) * 8;        // 0,8,16,24
      const float* src = A + (size_t)(rowBase + r) * arow + k0 + kk;
      float v[8];
      *(float4*)(v + 0) = *(const float4*)(src + 0);
      *(float4*)(v + 4) = *(const float4*)(src + 4);
      if (maskSalt) {
#pragma unroll
        for (int i = 0; i < 8; ++i)
          v[i] *= drop_mask(maskSalt, t, rowBase + r, k0 + kk + i);
      }
#pragma unroll
      for (int i = 0; i < 8; ++i) lA[r * LDA + kk + i] = v[i];
    }
    // ---- stage B chunk: 32 k x 128 n, K-pair interleaved ----
    // thread handles K-pair (2*kp, 2*kp+1), 8 columns: 4x global_load_b128,
    // 8x ds_store_b64 of {even,odd}.
    {
      int kp = tid >> 4;             // 0..15  (K pair index)
      int n  = (tid & 15) * 8;       // 0..120
      const float* se = W + (size_t)(k0 + 2 * kp + 0) * U_ + colBase + n;
      const float* so = W + (size_t)(k0 + 2 * kp + 1) * U_ + colBase + n;
      float ve[8], vo[8];
      *(float4*)(ve + 0) = *(const float4*)(se + 0);
      *(float4*)(ve + 4) = *(const float4*)(se + 4);
      *(float4*)(vo + 0) = *(const float4*)(so + 0);
      *(float4*)(vo + 4) = *(const float4*)(so + 4);
      float* dst = lB + kp * LDBP + 2 * n;
#pragma unroll
      for (int i = 0; i < 8; ++i) {
        v2f pair; pair[0] = ve[i]; pair[1] = vo[i];
        *(v2f*)(dst + 2 * i) = pair;
      }
    }
    __syncthreads();
    // ---- compute: 8 micro-steps of K=4, 4 WMMAs each (all b64 DS loads) ----
#pragma unroll
    for (int k4 = 0; k4 < KC; k4 += 4) {
      v2f af[2], bf[2];
#pragma unroll
      for (int i = 0; i < 2; ++i)   // A frag: M=lm, K=k4+lk2+{0,1}
        af[i] = *(const v2f*)&lA[(m0 + i * 16 + lm) * LDA + k4 + lk2];
#pragma unroll
      for (int j = 0; j < 2; ++j)   // B frag: one packed b64
        bf[j] = *(const v2f*)&lB[((k4 + lk2) >> 1) * LDBP +
                                 2 * (n0 + j * 16 + lm)];
#pragma unroll
      for (int i = 0; i < 2; ++i)
#pragma unroll
        for (int j = 0; j < 2; ++j)
          acc[i][j] = __builtin_amdgcn_wmma_f32_16x16x4_f32(
              false, af[i], false, bf[j], (short)0, acc[i][j], false, false);
    }
  }
}

// pre = (x*mx)@Wx + (s*ms)@Ws + o@Wo + b ; s_new = tanh(pre)
__global__ void __launch_bounds__(NTHR)
rnn_step_state(const float* __restrict__ x_t,      // x + t*D, row stride T*D
               const float* __restrict__ s_in,     // [B,U]
               const float* __restrict__ o_in,     // [B,O]
               const float* __restrict__ wx_t,     // [D,U]
               const float* __restrict__ ws_t,     // [U,U]
               const float* __restrict__ wo_t,     // [O,U]
               const float* __restrict__ b_t,      // [U]
               float* __restrict__ s_out,          // [B,U]
               int t) {
  __shared__ float lA[BM * LDA];
  __shared__ float lB[(KC / 2) * LDBP];
  const int lane = threadIdx.x & 31;
  const int wave = threadIdx.x >> 5;
  const int wr = wave & 1, wc = wave >> 1;
  const int rowBase = blockIdx.y * BM;
  const int colBase = blockIdx.x * BN;

  v8f acc[2][2] = {};
  gemm_segment(x_t,  (size_t)T_ * D_, wx_t, D_, rowBase, colBase, 1u, t,
               lA, lB, acc, wr, wc, lane);
  gemm_segment(s_in, (size_t)U_,      ws_t, U_, rowBase, colBase, 2u, t,
               lA, lB, acc, wr, wc, lane);
  gemm_segment(o_in, (size_t)O_,      wo_t, O_, rowBase, colBase, 0u, t,
               lA, lB, acc, wr, wc, lane);

  const int lm = lane & 15, hi = lane >> 4;
#pragma unroll
  for (int j = 0; j < 2; ++j) {
    int c = colBase + wc * 32 + j * 16 + lm;
    float bias = b_t[c];
#pragma unroll
    for (int i = 0; i < 2; ++i) {
      int rb = rowBase + wr * 32 + i * 16 + hi * 8;
#pragma unroll
      for (int v = 0; v < 8; ++v)
        s_out[(size_t)(rb + v) * U_ + c] = tanhf(acc[i][j][v] + bias);
    }
  }
}

// o_new = (s_new @ Wy + by) * mo ; write out[:,t,:] and o ping-pong buffer
__global__ void __launch_bounds__(NTHR)
rnn_step_out(const float* __restrict__ s_new,     // [B,U]
             const float* __restrict__ wy_t,      // [U,O]
             const float* __restrict__ by_t,      // [O]
             float* __restrict__ out_t,           // d_out + t*O, row stride T*O
             float* __restrict__ o_out,           // [B,O]
             int t) {
  __shared__ float lA[BM * LDA];
  __shared__ float lB[(KC / 2) * LDBP];
  const int lane = threadIdx.x & 31;
  const int wave = threadIdx.x >> 5;
  const int wr = wave & 1, wc = wave >> 1;
  const int rowBase = blockIdx.y * BM;
  const int colBase = blockIdx.x * BN;

  v8f acc[2][2] = {};
  gemm_segment(s_new, (size_t)U_, wy_t, U_, rowBase, colBase, 0u, t,
               lA, lB, acc, wr, wc, lane);

  const int lm = lane & 15, hi = lane >> 4;
#pragma unroll
  for (int j = 0; j < 2; ++j) {
    int c = colBase + wc * 32 + j * 16 + lm;
    float bias = by_t[c];
#pragma unroll
    for (int i = 0; i < 2; ++i) {
      int rb = rowBase + wr * 32 + i * 16 + hi * 8;
#pragma unroll
      for (int v = 0; v < 8; ++v) {
        int r = rb + v;
        float val = (acc[i][j][v] + bias) * drop_mask(3u, t, r, c);
        o_out[(size_t)r * O_ + c]        = val;
        out_t[(size_t)r * (T_ * O_) + c] = val;
      }
    }
  }
}

extern "C" void kernel_launch(void* const* d_in, const int* in_sizes, int n_in,
                              void* d_out, int out_size, void* d_ws, size_t ws_size,
                              hipStream_t stream) {
  const float* x   = (const float*)d_in[0];  // [B,T,D]
  const float* s0  = (const float*)d_in[1];  // [B,U]
  const float* o0  = (const float*)d_in[2];  // [B,O]
  const float* Wx  = (const float*)d_in[3];  // [T,D,U]
  const float* Ws  = (const float*)d_in[4];  // [T,U,U]
  const float* Wo  = (const float*)d_in[5];  // [T,O,U]
  const float* Wy  = (const float*)d_in[6];  // [T,U,O]
  const float* b   = (const float*)d_in[7];  // [T,U]
  const float* by  = (const float*)d_in[8];  // [T,O]
  float* out = (float*)d_out;

  float* ws = (float*)d_ws;
  const size_t SU = (size_t)B_ * U_;
  float* sbuf[2] = { ws,           ws + SU };
  float* obuf[2] = { ws + 2 * SU,  ws + 3 * SU };

  dim3 grid(U_ / BN, B_ / BM);   // 4 x 8 = 32 blocks
  dim3 blk(NTHR);

  for (int t = 0; t < T_; ++t) {
    const float* s_in = (t == 0) ? s0 : sbuf[(t + 1) & 1];
    const float* o_in = (t == 0) ? o0 : obuf[(t + 1) & 1];
    float* s_out = sbuf[t & 1];
    float* o_out = obuf[t & 1];

    rnn_step_state<<<grid, blk, 0, stream>>>(
        x + (size_t)t * D_, s_in, o_in,
        Wx + (size_t)t * D_ * U_,
        Ws + (size_t)t * U_ * U_,
        Wo + (size_t)t * O_ * U_,
        b + (size_t)t * U_,
        s_out, t);

    rnn_step_out<<<grid, blk, 0, stream>>>(
        s_out,
        Wy + (size_t)t * U_ * O_,
        by + (size_t)t * O_,
        out + (size_t)t * O_,
        o_out, t);
  }
}